// HighFidelityPerformanceModel_4939212390505
// MI455X (gfx1250) — compile-verified
//
#include <hip/hip_runtime.h>
#include <math.h>

typedef float v2f __attribute__((ext_vector_type(2)));
typedef float v8f __attribute__((ext_vector_type(8)));

#define TPB 256
#define MAX_BLK 1024

// Full 32-lane f32 sum using V_WMMA_F32_16X16X4_F32.
// Stage 1: A = per-lane value (A vgpr1 = 0), B = ones  => D[m][n] = v[m] + v[m+16]
// Stage 2: A = ones, B = pairs of stage-1 D VGPRs, accumulated over 4 WMMAs.
// Every (D-vgpr, lane-half) slot is consumed exactly once under sum_k B[k][n],
// so the result is sum over all 32 lanes, replicated to every lane.
__device__ __forceinline__ float wave_sum32(float v) {
  v2f a;    a.x = v;    a.y = 0.0f;
  v2f ones; ones.x = 1.0f; ones.y = 1.0f;
  v8f zero = {0.f, 0.f, 0.f, 0.f, 0.f, 0.f, 0.f, 0.f};
  v8f d1 = __builtin_amdgcn_wmma_f32_16x16x4_f32(
      false, a, false, ones, (short)0, zero, false, false);
  v8f acc = zero;
  v2f b;
  b.x = d1[0]; b.y = d1[1];
  acc = __builtin_amdgcn_wmma_f32_16x16x4_f32(false, ones, false, b, (short)0, acc, false, false);
  b.x = d1[2]; b.y = d1[3];
  acc = __builtin_amdgcn_wmma_f32_16x16x4_f32(false, ones, false, b, (short)0, acc, false, false);
  b.x = d1[4]; b.y = d1[5];
  acc = __builtin_amdgcn_wmma_f32_16x16x4_f32(false, ones, false, b, (short)0, acc, false, false);
  b.x = d1[6]; b.y = d1[7];
  acc = __builtin_amdgcn_wmma_f32_16x16x4_f32(false, ones, false, b, (short)0, acc, false, false);
  return acc[0];
}

__global__ void __launch_bounds__(TPB) pm_main(
    const float* __restrict__ T, const float* __restrict__ S,
    const float* __restrict__ num_pes, const float* __restrict__ bufs,
    float* __restrict__ partials, int G)
{
  const float pe = num_pes[0];
  const float sq = sqrtf(pe);
  const float b0 = bufs[0], b1 = bufs[1], b2 = bufs[2];
  const float onchip = 8.0f * sq;                       // 2*sq*4*1000e6/1e9
  const float inv_bw0 = 1.0f / (100.0f * 1e9f + 1e-9f); // DRAM interface
  const float inv_bw1 = 1.0f / (onchip * 1e9f + 1e-9f); // on-chip interfaces
  const float inv_comp = 1.0f / (pe * 1e9f + 1e-9f);    // pe * CLK_MHZ * 1e6
  const float epa0 = 1.2f  + 0.002f * b2;
  const float epa1 = 0.52f + 0.01f * (b1 / sq);
  const float epa2 = 0.18f;
  const float KB = 4.0f / 1024.0f;

  float lat_s = 0.0f, en_s = 0.0f, mis_s = 0.0f;

  const int tid = blockIdx.x * blockDim.x + threadIdx.x;
  const int stride = gridDim.x * blockDim.x;

  for (int g = tid; g < G; g += stride) {
    const float4* t4 = (const float4*)(T + (size_t)g * 28); // 112B, 16B-aligned
    const float4* s4 = (const float4*)(S + (size_t)g * 28);

    // Tensor types: 0=Input 1=Weight 2=Output; dims N,K,C,P,Q,R,S
    const bool DM[3][7] = {{1,0,1,1,1,0,0},{0,1,1,0,0,1,1},{1,1,0,1,1,0,0}};
    const bool RM[3][7] = {{0,1,0,0,0,0,0},{1,0,0,1,1,0,0},{0,0,1,0,0,1,1}};

    float fp[3][4], op[3][4], ru[3][4];
#pragma unroll
    for (int t = 0; t < 3; ++t)
#pragma unroll
      for (int l = 0; l < 4; ++l) { fp[t][l] = 1.f; op[t][l] = 1.f; ru[t][l] = 1.f; }
    float macs = 1.0f;

#pragma unroll
    for (int d = 0; d < 7; ++d) {
      const float4 t = t4[d];
      const float4 s = s4[d];
      float ct[4], cts[4], st[4];
      ct[0] = t.x; ct[1] = ct[0] * t.y; ct[2] = ct[1] * t.z; ct[3] = ct[2] * t.w;
      const float p0 = t.x * s.x, p1 = t.y * s.y, p2 = t.z * s.z, p3 = t.w * s.w;
      cts[0] = p0; cts[1] = cts[0] * p1; cts[2] = cts[1] * p2; cts[3] = cts[2] * p3;
      st[3] = t.w; st[2] = t.z * st[3]; st[1] = t.y * st[2]; st[0] = t.x * st[1];
      macs *= cts[3];
#pragma unroll
      for (int tt = 0; tt < 3; ++tt) {
        if (DM[tt][d]) {
#pragma unroll
          for (int l = 0; l < 4; ++l) { fp[tt][l] *= cts[l]; op[tt][l] *= st[l]; }
        }
        if (RM[tt][d]) {
#pragma unroll
          for (int l = 0; l < 4; ++l) ru[tt][l] *= ct[l];
        }
      }
    }

    // interfaces keyed by lower level: low = {2,1,0}, up = {3,2,1}
    float acc0 = 0.f, acc1 = 0.f, acc2 = 0.f;
#pragma unroll
    for (int tt = 0; tt < 3; ++tt) {
      acc0 += fp[tt][2] * (op[tt][3] / (ru[tt][2] + 1e-9f));
      acc1 += fp[tt][1] * (op[tt][2] / (ru[tt][1] + 1e-9f));
      acc2 += fp[tt][0] * (op[tt][1] / (ru[tt][0] + 1e-9f));
    }
    acc0 *= 4.0f; acc1 *= 4.0f; acc2 *= 4.0f;

    const float mem_lat = fmaxf(acc0 * inv_bw0, fmaxf(acc1 * inv_bw1, acc2 * inv_bw1));
    lat_s += fmaxf(macs * inv_comp, mem_lat);
    en_s  += macs * 0.56f + 0.25f * (acc0 * epa0 + acc1 * epa1 + acc2 * epa2);

    float r, dfc;
    r = (fp[0][0] + fp[1][0] + fp[2][0]) * KB; dfc = fmaxf(r - b0, 0.f); mis_s += dfc * dfc;
    r = (fp[0][1] + fp[1][1] + fp[2][1]) * KB; dfc = fmaxf(r - b1, 0.f); mis_s += dfc * dfc;
    r = (fp[0][2] + fp[1][2] + fp[2][2]) * KB; dfc = fmaxf(r - b2, 0.f); mis_s += dfc * dfc;
  }

  // Wave-level reduction via WMMA (EXEC is all-ones here: full block, converged).
  lat_s = wave_sum32(lat_s);
  en_s  = wave_sum32(en_s);
  mis_s = wave_sum32(mis_s);

  __shared__ float sw[TPB / 32][3];
  const int w = threadIdx.x >> 5;
  if ((threadIdx.x & 31) == 0) { sw[w][0] = lat_s; sw[w][1] = en_s; sw[w][2] = mis_s; }
  __syncthreads();
  if (threadIdx.x == 0) {
    float a = 0.f, b = 0.f, c = 0.f;
#pragma unroll
    for (int i = 0; i < TPB / 32; ++i) { a += sw[i][0]; b += sw[i][1]; c += sw[i][2]; }
    partials[blockIdx.x * 3 + 0] = a;
    partials[blockIdx.x * 3 + 1] = b;
    partials[blockIdx.x * 3 + 2] = c;
  }
}

__global__ void __launch_bounds__(32) pm_final(
    const float* __restrict__ partials, const float* __restrict__ num_pes,
    const float* __restrict__ bufs, float* __restrict__ out, int nblk)
{
  const int lane = threadIdx.x;
  float l0 = 0.f, l1 = 0.f, l2 = 0.f;
  for (int i = lane; i < nblk; i += 32) {
    l0 += partials[i * 3 + 0];
    l1 += partials[i * 3 + 1];
    l2 += partials[i * 3 + 2];
  }
  l0 = wave_sum32(l0);
  l1 = wave_sum32(l1);
  l2 = wave_sum32(l2);
  if (lane == 0) {
    const float pe = num_pes[0];
    const float area = pe * 0.01f + (bufs[0] + bufs[1] + bufs[2]) * 0.005f;
    out[0] = l0;   // total_latency
    out[1] = l1;   // total_energy
    out[2] = area; // area
    out[3] = l2;   // total_mismatch
  }
}

extern "C" void kernel_launch(void* const* d_in, const int* in_sizes, int n_in,
                              void* d_out, int out_size, void* d_ws, size_t ws_size,
                              hipStream_t stream) {
  const float* T    = (const float*)d_in[0]; // temporal_factors [G,7,4]
  const float* S    = (const float*)d_in[1]; // spatial_factors  [G,7,4]
  const float* pe   = (const float*)d_in[2]; // num_pes [1]
  const float* bufs = (const float*)d_in[3]; // buffer_sizes_kb [3]
  const int G = in_sizes[0] / 28;

  int nblk = MAX_BLK;
  const size_t need = (size_t)nblk * 3 * sizeof(float);
  if (need > ws_size) nblk = (int)(ws_size / (3 * sizeof(float)));
  if (nblk < 1) nblk = 1;

  float* partials = (float*)d_ws;
  pm_main<<<nblk, TPB, 0, stream>>>(T, S, pe, bufs, partials, G);
  pm_final<<<1, 32, 0, stream>>>(partials, pe, bufs, (float*)d_out, nblk);
}